// SphericalCriterion_75316546503349
// MI455X (gfx1250) — compile-verified
//
#include <hip/hip_runtime.h>

typedef __attribute__((ext_vector_type(2))) float v2f;
typedef __attribute__((ext_vector_type(4))) float v4f;
typedef __attribute__((ext_vector_type(8))) float v8f;

#define NROWS 4096
#define NCOLS 6144
#define NB    1024   // partial-reduction blocks (must be 64 * 16 for the WMMA finisher)
#define NT    256    // threads per block (8 waves on wave32)

// Process one f32 element: s = sigmoid(p); accumulate into the three loss partials.
__device__ __forceinline__ void accum_elem(float p, float t, int rowmod, int colmod3, int col,
                                           float& r, float& th, float& ph) {
    float s = 1.0f / (1.0f + expf(-p));
    float a = fabsf(s - t);
    r  += (rowmod == 0) ? a : 0.0f;
    th += (rowmod == 1) ? a : 0.0f;
    float d = fminf(a, fminf(fabsf(s - (t + 1.0f)), fabsf(s - (t - 1.0f))));
    ph += ((colmod3 == 2) && (col < NROWS)) ? d : 0.0f;   // v_idx = arange(2, 4096, 3) on columns
}

// Pass 1: streaming, bandwidth-bound. float4 non-temporal loads, branch-free predication,
// deterministic block-local reduction, one 3-float partial per block.
__global__ void spherical_partial(const float* __restrict__ preds,
                                  const float* __restrict__ targs,
                                  float* __restrict__ partial) {
    const int vecsPerRow = NCOLS / 4;                 // 1536, row-aligned float4s
    const int totalVec   = NROWS * vecsPerRow;        // 6,291,456
    float r = 0.f, th = 0.f, ph = 0.f;

    for (int v = blockIdx.x * blockDim.x + threadIdx.x; v < totalVec;
         v += gridDim.x * blockDim.x) {
        int row    = v / vecsPerRow;
        int c4     = v - row * vecsPerRow;
        int col    = c4 * 4;
        int rowmod = row % 3;
        int cm     = c4 % 3;                          // (4*c4) % 3 == c4 % 3

        v4f p = __builtin_nontemporal_load((const v4f*)preds + v);
        v4f t = __builtin_nontemporal_load((const v4f*)targs + v);

        #pragma unroll
        for (int i = 0; i < 4; ++i) {
            int cmi = cm + i; if (cmi >= 3) cmi -= 3;
            accum_elem(p[i], t[i], rowmod, cmi, col + i, r, th, ph);
        }
    }

    // wave32 shuffle tree
    #pragma unroll
    for (int off = 16; off > 0; off >>= 1) {
        r  += __shfl_down(r,  off, 32);
        th += __shfl_down(th, off, 32);
        ph += __shfl_down(ph, off, 32);
    }

    __shared__ float red[3][NT / 32];
    int wave = threadIdx.x >> 5;
    int lane = threadIdx.x & 31;
    if (lane == 0) { red[0][wave] = r; red[1][wave] = th; red[2][wave] = ph; }
    __syncthreads();
    if (threadIdx.x == 0) {
        float R = 0.f, T = 0.f, P = 0.f;
        #pragma unroll
        for (int w = 0; w < NT / 32; ++w) { R += red[0][w]; T += red[1][w]; P += red[2][w]; }
        partial[blockIdx.x * 3 + 0] = R;
        partial[blockIdx.x * 3 + 1] = T;
        partial[blockIdx.x * 3 + 2] = P;
    }
}

// Pass 2: one wave. Reduce the 1024 block-partials per component with
// V_WMMA_F32_16X16X4_F32 using an all-ones B matrix: D = A*ones + C sums each
// A tile's K dimension; 16 chained tiles consume all 1024 values. The D matrix
// (8 VGPRs) is then collapsed with 7 adds + one lane^16 shuffle (rows 0-7 sit
// in lanes 0-15, rows 8-15 in lanes 16-31, N replicated). EXEC is all-1s.
__global__ void spherical_final(const float* __restrict__ partial,
                                float* __restrict__ out) {
    int lane = threadIdx.x;   // 32 threads, no divergence before WMMA
    float sums[3];

    v2f bones; bones[0] = 1.0f; bones[1] = 1.0f;

    #pragma unroll
    for (int comp = 0; comp < 3; ++comp) {
        v8f acc = {};
        #pragma unroll
        for (int chunk = 0; chunk < NB / 64; ++chunk) {
            int i0 = chunk * 64 + lane * 2;           // two A values per lane per tile
            v2f a;
            a[0] = partial[(i0    ) * 3 + comp];
            a[1] = partial[(i0 + 1) * 3 + comp];
            acc = __builtin_amdgcn_wmma_f32_16x16x4_f32(
                false, a, false, bones, (short)0, acc, false, false);
        }
        float s = acc[0] + acc[1] + acc[2] + acc[3] + acc[4] + acc[5] + acc[6] + acc[7];
        s += __shfl_xor(s, 16, 32);                   // combine M=0..7 half with M=8..15 half
        sums[comp] = s;
    }

    if (lane == 0) {
        float r  = sums[0] / (1366.0f * 6144.0f);     // rows 0::3 of 4096 -> 1366 rows
        float th = sums[1] / (1365.0f * 6144.0f);     // rows 1::3 -> 1365 rows
        float ph = sums[2] / (4096.0f * 2048.0f);     // reference divides by b * n_vecs
        out[0] = r + th + ph;
        out[1] = r;
        out[2] = th;
        out[3] = ph;
    }
}

extern "C" void kernel_launch(void* const* d_in, const int* in_sizes, int n_in,
                              void* d_out, int out_size, void* d_ws, size_t ws_size,
                              hipStream_t stream) {
    const float* preds = (const float*)d_in[0];
    const float* targs = (const float*)d_in[1];
    float* partial     = (float*)d_ws;               // NB * 3 floats = 12 KB, fully rewritten each call
    float* out         = (float*)d_out;

    spherical_partial<<<NB, NT, 0, stream>>>(preds, targs, partial);
    spherical_final<<<1, 32, 0, stream>>>(partial, out);
}